// GIN_23476291240660
// MI455X (gfx1250) — compile-verified
//
#include <hip/hip_runtime.h>
#include <math.h>

typedef __attribute__((ext_vector_type(2))) float v2f;
typedef __attribute__((ext_vector_type(8))) float v8f;

#define N_NODES 100000
#define N_EDGES 3200000
#define N_GRAPHS 256
#define H 90
#define HP 96 // padded feature dim (6 WMMA tiles of 16)

// ---------------- utility kernels ----------------

__global__ void fill_zero_kernel(float* __restrict__ p, int n) {
  int i = blockIdx.x * blockDim.x + threadIdx.x;
  int stride = gridDim.x * blockDim.x;
  for (; i < n; i += stride) p[i] = 0.f;
}

// Pad [H x H] weight into [HP x HP] (zeros outside), bias H -> HP.
__global__ void pad_weight_kernel(const float* __restrict__ W, const float* __restrict__ b,
                                  float* __restrict__ Wp, float* __restrict__ bp) {
  int idx = blockIdx.x * blockDim.x + threadIdx.x;
  if (idx < HP * HP) {
    int n = idx / HP, k = idx - n * HP;
    Wp[idx] = (n < H && k < H) ? W[n * H + k] : 0.f;
  }
  if (idx < HP) bp[idx] = (idx < H) ? b[idx] : 0.f;
}

// x [N_NODES x H] -> h0 [N_NODES x HP], pad columns zeroed.
__global__ void copy_pad_x_kernel(const float* __restrict__ x, float* __restrict__ h0) {
  int i = blockIdx.x * blockDim.x + threadIdx.x;
  int stride = gridDim.x * blockDim.x;
  const int total = N_NODES * HP;
  for (; i < total; i += stride) {
    int node = i / HP, f = i - node * HP;
    h0[i] = (f < H) ? x[node * H + f] : 0.f;
  }
}

// One wave32 per edge: agg[dst] += h[src] (coalesced row gather + L2 atomics).
__global__ void scatter_kernel(const float* __restrict__ h, float* __restrict__ agg,
                               const long long* __restrict__ src,
                               const long long* __restrict__ dst) {
  int gtid = blockIdx.x * blockDim.x + threadIdx.x;
  int edge = gtid >> 5;
  int lane = gtid & 31;
  if (edge >= N_EDGES) return;
  long long s = src[edge];
  long long d = dst[edge];
  const float* hs = h + (size_t)s * HP;
  float* ad = agg + (size_t)d * HP;
  for (int f = lane; f < H; f += 32)
    atomicAdd(&ad[f], hs[f]);
}

// Fused GINConv linear: hout = ELU((hin + agg) @ Wp^T + bp), via fp32 WMMA 16x16x4.
// Block = (32, 6): 6 waves, each owns one 16-wide N tile of a 16-row M tile.
__global__ __launch_bounds__(192)
void gin_gemm_elu_wmma_kernel(const float* __restrict__ hin, const float* __restrict__ agg,
                              const float* __restrict__ Wp, const float* __restrict__ bp,
                              float* __restrict__ hout) {
  const int lane  = threadIdx.x;          // 0..31
  const int row0  = blockIdx.x * 16;      // M tile base (node rows)
  const int n0    = threadIdx.y * 16;     // N tile base (output features)
  const int l15   = lane & 15;
  const int khalf = (lane >> 4) << 1;     // lanes 0-15 hold K=0,1; lanes 16-31 hold K=2,3

  v8f acc = {0.f, 0.f, 0.f, 0.f, 0.f, 0.f, 0.f, 0.f};

  const float* arow = hin + (size_t)(row0 + l15) * HP + khalf;
  const float* grow = agg + (size_t)(row0 + l15) * HP + khalf;
  const float* brow = Wp  + (size_t)(n0   + l15) * HP + khalf; // B[k,n] = Wp[n*HP + k]

#pragma unroll 4
  for (int k0 = 0; k0 < HP; k0 += 4) {
    v2f a, b;
    a.x = arow[k0]     + grow[k0];
    a.y = arow[k0 + 1] + grow[k0 + 1];
    b.x = brow[k0];
    b.y = brow[k0 + 1];
    // 8 args: (neg_a, A, neg_b, B, c_mod, C, reuse_a, reuse_b)
    acc = __builtin_amdgcn_wmma_f32_16x16x4_f32(false, a, false, b, (short)0, acc,
                                                false, false);
  }

  // D layout: VGPR r -> lanes 0-15: (M=r, N=lane); lanes 16-31: (M=8+r, N=lane-16)
  const int mbase = row0 + ((lane >> 4) << 3);
  const int nc = n0 + l15;
  const float bias = bp[nc];
#pragma unroll
  for (int r = 0; r < 8; ++r) {
    float v = acc[r] + bias;
    v = (v > 0.f) ? v : expm1f(v); // ELU(alpha=1)
    hout[(size_t)(mbase + r) * HP + nc] = v;
  }
}

// One wave32 per node: sums[batch[i]] += h3[i], counts[batch[i]] += 1
__global__ void pool_kernel(const float* __restrict__ h3, const long long* __restrict__ batch,
                            float* __restrict__ sums, float* __restrict__ counts) {
  int gtid = blockIdx.x * blockDim.x + threadIdx.x;
  int node = gtid >> 5;
  int lane = gtid & 31;
  if (node >= N_NODES) return;
  long long g = batch[node];
  const float* hr = h3 + (size_t)node * HP;
  float* sr = sums + (size_t)g * HP;
  for (int f = lane; f < H; f += 32)
    atomicAdd(&sr[f], hr[f]);
  if (lane == 0) atomicAdd(&counts[g], 1.f);
}

// One wave32 per graph: mean -> fc1(32) -> ELU -> fc2(1) via wave reduction.
__global__ void head_kernel(const float* __restrict__ sums, const float* __restrict__ counts,
                            const float* __restrict__ Wf1, const float* __restrict__ bf1,
                            const float* __restrict__ Wf2, const float* __restrict__ bf2,
                            float* __restrict__ out) {
  int g = blockIdx.x;
  int j = threadIdx.x; // 0..31, one fc1 output per lane
  float inv = 1.f / fmaxf(counts[g], 1.f);
  float acc = bf1[j];
  const float* srow = sums + (size_t)g * HP;
  const float* wrow = Wf1 + j * H;
  for (int k = 0; k < H; ++k)
    acc = fmaf(srow[k] * inv, wrow[k], acc);
  float o = (acc > 0.f) ? acc : expm1f(acc);
  float t = o * Wf2[j];
#pragma unroll
  for (int off = 16; off > 0; off >>= 1)
    t += __shfl_xor(t, off, 32);
  if (j == 0) out[g] = t + bf2[0];
}

// ---------------- launch ----------------

extern "C" void kernel_launch(void* const* d_in, const int* in_sizes, int n_in,
                              void* d_out, int out_size, void* d_ws, size_t ws_size,
                              hipStream_t stream) {
  const float*      x     = (const float*)d_in[0];
  const long long*  ei    = (const long long*)d_in[1]; // int64 [2, N_EDGES]
  const long long*  batch = (const long long*)d_in[2]; // int64 [N_NODES]
  const float* W1  = (const float*)d_in[3];  const float* b1  = (const float*)d_in[4];
  const float* W2  = (const float*)d_in[5];  const float* b2  = (const float*)d_in[6];
  const float* W3  = (const float*)d_in[7];  const float* b3  = (const float*)d_in[8];
  const float* Wf1 = (const float*)d_in[9];  const float* bf1 = (const float*)d_in[10];
  const float* Wf2 = (const float*)d_in[11]; const float* bf2 = (const float*)d_in[12];
  float* out = (float*)d_out;

  const long long* src = ei;
  const long long* dst = ei + N_EDGES;

  // workspace carve-up (floats)
  const size_t nodeBuf = (size_t)N_NODES * HP;
  float* ws  = (float*)d_ws;
  float* hA  = ws;  ws += nodeBuf;
  float* hB  = ws;  ws += nodeBuf;
  float* agg = ws;  ws += nodeBuf;
  float* Wp1 = ws;  ws += HP * HP;
  float* Wp2 = ws;  ws += HP * HP;
  float* Wp3 = ws;  ws += HP * HP;
  float* bp1 = ws;  ws += HP;
  float* bp2 = ws;  ws += HP;
  float* bp3 = ws;  ws += HP;
  float* sums = ws; ws += (size_t)N_GRAPHS * HP;
  float* cnts = ws; ws += N_GRAPHS;
  if ((size_t)((char*)ws - (char*)d_ws) > ws_size) return; // not enough scratch

  const dim3 padGrd((HP * HP + 255) / 256), blk256(256);
  pad_weight_kernel<<<padGrd, blk256, 0, stream>>>(W1, b1, Wp1, bp1);
  pad_weight_kernel<<<padGrd, blk256, 0, stream>>>(W2, b2, Wp2, bp2);
  pad_weight_kernel<<<padGrd, blk256, 0, stream>>>(W3, b3, Wp3, bp3);

  copy_pad_x_kernel<<<2048, blk256, 0, stream>>>(x, hA);

  const dim3 scatGrd((unsigned)((N_EDGES * 32 + 255) / 256));
  const dim3 gemmGrd(N_NODES / 16);
  const dim3 gemmBlk(32, HP / 16);
  const int aggN = (int)nodeBuf;

  // layer 1: hB = ELU((hA + agg(hA)) @ W1^T + b1)
  fill_zero_kernel<<<2048, blk256, 0, stream>>>(agg, aggN);
  scatter_kernel<<<scatGrd, blk256, 0, stream>>>(hA, agg, src, dst);
  gin_gemm_elu_wmma_kernel<<<gemmGrd, gemmBlk, 0, stream>>>(hA, agg, Wp1, bp1, hB);

  // layer 2: hA = ELU((hB + agg(hB)) @ W2^T + b2)
  fill_zero_kernel<<<2048, blk256, 0, stream>>>(agg, aggN);
  scatter_kernel<<<scatGrd, blk256, 0, stream>>>(hB, agg, src, dst);
  gin_gemm_elu_wmma_kernel<<<gemmGrd, gemmBlk, 0, stream>>>(hB, agg, Wp2, bp2, hA);

  // layer 3: hB = ELU((hA + agg(hA)) @ W3^T + b3)
  fill_zero_kernel<<<2048, blk256, 0, stream>>>(agg, aggN);
  scatter_kernel<<<scatGrd, blk256, 0, stream>>>(hA, agg, src, dst);
  gin_gemm_elu_wmma_kernel<<<gemmGrd, gemmBlk, 0, stream>>>(hA, agg, Wp3, bp3, hB);

  // global mean pool + MLP head
  fill_zero_kernel<<<64, blk256, 0, stream>>>(sums, N_GRAPHS * HP);
  fill_zero_kernel<<<1, blk256, 0, stream>>>(cnts, N_GRAPHS);
  pool_kernel<<<(N_NODES * 32 + 255) / 256, blk256, 0, stream>>>(hB, batch, sums, cnts);
  head_kernel<<<N_GRAPHS, 32, 0, stream>>>(sums, cnts, Wf1, bf1, Wf2, bf2, out);
}